// L2PairwiceObjectiveFunction_33328946217784
// MI455X (gfx1250) — compile-verified
//
#include <hip/hip_runtime.h>
#include <hip/hip_bf16.h>
#include <math.h>

#define B_DIM  1024
#define N_DIM  8192
#define M_DIM  3000
#define MP_DIM 3008   // M padded to multiple of 4 (WMMA K-step), zeros contribute nothing
#define EPSV   1e-8f

typedef float v2f __attribute__((ext_vector_type(2)));
typedef float v8f __attribute__((ext_vector_type(8)));

// ---------------- Kernel 1: scalar min/max over row endpoints ----------------
__global__ void __launch_bounds__(1024)
minmax_kernel(const float* __restrict__ x, float* __restrict__ mm) {
    __shared__ float smin[1024];
    __shared__ float smax[1024];
    int t = threadIdx.x;
    smin[t] = x[(size_t)t * N_DIM];                 // x[t, 0]
    smax[t] = x[(size_t)t * N_DIM + (N_DIM - 1)];   // x[t, N-1]
    __syncthreads();
    for (int s = 512; s > 0; s >>= 1) {
        if (t < s) {
            smin[t] = fminf(smin[t], smin[t + s]);
            smax[t] = fmaxf(smax[t], smax[t + s]);
        }
        __syncthreads();
    }
    if (t == 0) { mm[0] = smin[0]; mm[1] = smax[0]; }
}

// ---------------- Kernel 2: per-row interpolation onto common grid ----------------
// One block per row. x row (32KB) cached in LDS for the 13-step binary search;
// y gathers hit a hot 32KB global row (L0/L2 resident).
__global__ void __launch_bounds__(256)
interp_kernel(const float* __restrict__ x, const float* __restrict__ y1,
              const float* __restrict__ y2, const float* __restrict__ mm,
              float* __restrict__ y1c, float* __restrict__ y2c,
              float* __restrict__ sq1, float* __restrict__ sq2) {
    __shared__ float4 xs4[N_DIM / 4];
    __shared__ float  red1[256];
    __shared__ float  red2[256];
    float* xs = (float*)xs4;

    const int b   = blockIdx.x;
    const int tid = threadIdx.x;
    const float4* xg = (const float4*)(x + (size_t)b * N_DIM);
    for (int i = tid; i < N_DIM / 4; i += 256) xs4[i] = xg[i];
    __syncthreads();

    const float* y1r = y1 + (size_t)b * N_DIM;
    const float* y2r = y2 + (size_t)b * N_DIM;

    const float xmin = mm[0];
    const float range = mm[1] - xmin;
    const float row_lo = xs[0];
    const float row_hi = xs[N_DIM - 1];

    float s1 = 0.f, s2 = 0.f;
    for (int m = tid; m < MP_DIM; m += 256) {
        float v1 = 0.f, v2 = 0.f;
        if (m < M_DIM) {
            float xc = xmin + ((float)m * (1.0f / (float)(M_DIM - 1))) * range;
            // searchsorted side='left' : first idx with xs[idx] >= xc
            int lo = 0, hi = N_DIM;
            #pragma unroll 1
            while (lo < hi) {
                int mid = (lo + hi) >> 1;
                if (xs[mid] < xc) lo = mid + 1; else hi = mid;
            }
            int i_lo = lo - 1;
            if (i_lo < 0) i_lo = 0;
            if (i_lo > N_DIM - 2) i_lo = N_DIM - 2;
            int i_hi = lo;
            if (i_hi > N_DIM - 1) i_hi = N_DIM - 1;

            float xl = xs[i_lo], xh = xs[i_hi];
            float denom = xh - xl;
            if (denom == 0.f) denom = 1.f;
            float w = (xc - xl) / (denom + 1e-9f);
            w = fminf(fmaxf(w, 0.f), 1.f);
            bool in = (xc >= row_lo) && (xc <= row_hi);

            float a1 = y1r[i_lo], b1 = y1r[i_hi];
            float a2 = y2r[i_lo], b2 = y2r[i_hi];
            v1 = in ? (a1 + w * (b1 - a1)) : 0.f;
            v2 = in ? (a2 + w * (b2 - a2)) : 0.f;
        }
        y1c[(size_t)b * MP_DIM + m] = v1;   // m in [3000,3008) padded with zeros
        y2c[(size_t)b * MP_DIM + m] = v2;
        s1 += v1 * v1;
        s2 += v2 * v2;
    }
    red1[tid] = s1; red2[tid] = s2;
    __syncthreads();
    for (int s = 128; s > 0; s >>= 1) {
        if (tid < s) { red1[tid] += red1[tid + s]; red2[tid] += red2[tid + s]; }
        __syncthreads();
    }
    if (tid == 0) {
        sq1[b] = red1[0] * (1.0f / (float)M_DIM);
        sq2[b] = red2[0] * (1.0f / (float)M_DIM);
    }
}

// ---------------- Kernel 3: fp32 WMMA GEMM + epilogue ----------------
// cross[i,j] = (1/M) * sum_k y1c[i,k]*y2c[j,k]; out = sqrt(2*max(sq1_i+sq2_j-2c,0)/(sq1_i+sq2_i+eps))
// Block = 4 waves; wave w computes rows [i0+w*16, +16) x cols [j0, j0+64).
__global__ void __launch_bounds__(128)
gemm_out_kernel(const float* __restrict__ y1c, const float* __restrict__ y2c,
                const float* __restrict__ sq1, const float* __restrict__ sq2,
                float* __restrict__ out) {
    const int wave = threadIdx.x >> 5;
    const int lane = threadIdx.x & 31;
    const int half = lane >> 4;      // 0: lanes 0-15, 1: lanes 16-31
    const int l16  = lane & 15;
    const int i0 = blockIdx.y * 64 + wave * 16;
    const int j0 = blockIdx.x * 64;

    // A 16x4 fp32 fragment layout: lanes 0-15 hold K=(0,1), lanes 16-31 hold K=(2,3)
    const float* ap  = y1c + (size_t)(i0 +      l16) * MP_DIM + 2 * half;
    const float* b0p = y2c + (size_t)(j0 +  0 + l16) * MP_DIM + 2 * half;
    const float* b1p = y2c + (size_t)(j0 + 16 + l16) * MP_DIM + 2 * half;
    const float* b2p = y2c + (size_t)(j0 + 32 + l16) * MP_DIM + 2 * half;
    const float* b3p = y2c + (size_t)(j0 + 48 + l16) * MP_DIM + 2 * half;

    v8f acc0 = {}, acc1 = {}, acc2 = {}, acc3 = {};
    #pragma unroll 4
    for (int k = 0; k < MP_DIM; k += 4) {
        v2f a  = *(const v2f*)(ap  + k);
        v2f b0 = *(const v2f*)(b0p + k);
        v2f b1 = *(const v2f*)(b1p + k);
        v2f b2 = *(const v2f*)(b2p + k);
        v2f b3 = *(const v2f*)(b3p + k);
        acc0 = __builtin_amdgcn_wmma_f32_16x16x4_f32(false, a, false, b0, (short)0, acc0, false, false);
        acc1 = __builtin_amdgcn_wmma_f32_16x16x4_f32(false, a, false, b1, (short)0, acc1, false, false);
        acc2 = __builtin_amdgcn_wmma_f32_16x16x4_f32(false, a, false, b2, (short)0, acc2, false, false);
        acc3 = __builtin_amdgcn_wmma_f32_16x16x4_f32(false, a, false, b3, (short)0, acc3, false, false);
    }

    const float inv_m = 1.0f / (float)M_DIM;
    v8f accs[4] = {acc0, acc1, acc2, acc3};

    float sq2j[4];
    #pragma unroll
    for (int t = 0; t < 4; ++t) sq2j[t] = sq2[j0 + t * 16 + l16];

    // C/D layout: VGPR r holds row r (lanes 0-15) / row 8+r (lanes 16-31), col = lane&15
    #pragma unroll
    for (int r = 0; r < 8; ++r) {
        int i = i0 + r + 8 * half;
        float s1   = sq1[i];
        float invb = 1.0f / (s1 + sq2[i] + EPSV);
        #pragma unroll
        for (int t = 0; t < 4; ++t) {
            int j = j0 + t * 16 + l16;
            float cross = accs[t][r] * inv_m;
            float diff  = fmaxf(s1 + sq2j[t] - 2.0f * cross, 0.f);
            out[(size_t)i * B_DIM + j] = sqrtf(2.0f * diff * invb);
        }
    }
}

extern "C" void kernel_launch(void* const* d_in, const int* in_sizes, int n_in,
                              void* d_out, int out_size, void* d_ws, size_t ws_size,
                              hipStream_t stream) {
    (void)in_sizes; (void)n_in; (void)out_size; (void)ws_size;
    const float* x  = (const float*)d_in[0];
    const float* y1 = (const float*)d_in[1];
    const float* y2 = (const float*)d_in[2];
    float* out = (float*)d_out;

    char* ws = (char*)d_ws;
    float* mm  = (float*)(ws);                  // 2 floats (padded to 256B)
    float* sq1 = (float*)(ws + 256);            // 1024 floats
    float* sq2 = (float*)(ws + 256 + 4096);     // 1024 floats
    float* y1c = (float*)(ws + 256 + 8192);     // 1024*3008 floats
    float* y2c = y1c + (size_t)B_DIM * MP_DIM;  // 1024*3008 floats

    minmax_kernel<<<1, 1024, 0, stream>>>(x, mm);
    interp_kernel<<<B_DIM, 256, 0, stream>>>(x, y1, y2, mm, y1c, y2c, sq1, sq2);
    dim3 grid(B_DIM / 64, B_DIM / 64);
    gemm_out_kernel<<<grid, 128, 0, stream>>>(y1c, y2c, sq1, sq2, out);
}